// GNNLayer_31138512896530
// MI455X (gfx1250) — compile-verified
//
#include <hip/hip_runtime.h>

#define DIM 128
#define LDSS 132   // padded LDS row stride (floats) to avoid bank conflicts

typedef __attribute__((ext_vector_type(2))) float v2f;
typedef __attribute__((ext_vector_type(8))) float v8f;

__device__ __forceinline__ float wave_sum(float v) {
#pragma unroll
    for (int m = 16; m >= 1; m >>= 1) v += __shfl_xor(v, m, 32);
    return v;
}

// ---------------------------------------------------------------------------
// Kernel 1: dual layer-norm. One wave32 per node, 4 channels per lane.
// ---------------------------------------------------------------------------
__global__ void __launch_bounds__(256) ln_kernel(
    const float* __restrict__ x,
    const float* __restrict__ gamma,   const float* __restrict__ beta,
    const float* __restrict__ gamma_r, const float* __restrict__ beta_r,
    float* __restrict__ h_f, float* __restrict__ h_r, int n_nodes) {
    const int wave = (int)(blockIdx.x * 8u) + (int)(threadIdx.x >> 5);
    const int lane = threadIdx.x & 31;
    if (wave >= n_nodes) return;

    const float4 v = ((const float4*)(x + (size_t)wave * DIM))[lane];
    float mu = wave_sum(v.x + v.y + v.z + v.w) * (1.0f / DIM);
    const float d0 = v.x - mu, d1 = v.y - mu, d2 = v.z - mu, d3 = v.w - mu;
    float var = wave_sum(d0 * d0 + d1 * d1 + d2 * d2 + d3 * d3) * (1.0f / DIM);
    const float inv = rsqrtf(var + 1e-5f);

    const float4 g  = ((const float4*)gamma)[lane];
    const float4 be = ((const float4*)beta)[lane];
    const float4 gr = ((const float4*)gamma_r)[lane];
    const float4 br = ((const float4*)beta_r)[lane];

    float4 o;
    o.x = d0 * inv * g.x + be.x;  o.y = d1 * inv * g.y + be.y;
    o.z = d2 * inv * g.z + be.z;  o.w = d3 * inv * g.w + be.w;
    ((float4*)(h_f + (size_t)wave * DIM))[lane] = o;
    o.x = d0 * inv * gr.x + br.x; o.y = d1 * inv * gr.y + br.y;
    o.z = d2 * inv * gr.z + br.z; o.w = d3 * inv * gr.w + br.w;
    ((float4*)(h_r + (size_t)wave * DIM))[lane] = o;
}

// ---------------------------------------------------------------------------
// Kernel 2: edge gather + fp32 atomic scatter-add, both directions in one pass.
// One wave32 per edge; lane owns 4 contiguous channels (coalesced float4).
// Accumulators (~103 MB) are L2-resident (192 MB L2) so global_atomic_add_f32
// resolves near line rate.
// ---------------------------------------------------------------------------
__device__ __forceinline__ void atomic_add_f32(float* p, float v) {
    __hip_atomic_fetch_add(p, v, __ATOMIC_RELAXED, __HIP_MEMORY_SCOPE_AGENT);
}

__global__ void __launch_bounds__(256) edge_kernel(
    const int* __restrict__ src, const int* __restrict__ dst,
    const float* __restrict__ h_f, const float* __restrict__ h_r,
    float* __restrict__ agg_f, float* __restrict__ agg_r,
    float* __restrict__ deg_f, float* __restrict__ deg_r, int n_edges) {
    const int wave = (int)(blockIdx.x * 8u) + (int)(threadIdx.x >> 5);
    const int lane = threadIdx.x & 31;
    if (wave >= n_edges) return;

    const int s = src[wave];
    const int d = dst[wave];
    {   // forward: h_f[src] -> agg_f[dst]
        const float4 m = ((const float4*)(h_f + (size_t)s * DIM))[lane];
        float* o = agg_f + (size_t)d * DIM + lane * 4;
        atomic_add_f32(o + 0, m.x); atomic_add_f32(o + 1, m.y);
        atomic_add_f32(o + 2, m.z); atomic_add_f32(o + 3, m.w);
    }
    {   // reverse: h_r[dst] -> agg_r[src]
        const float4 m = ((const float4*)(h_r + (size_t)d * DIM))[lane];
        float* o = agg_r + (size_t)s * DIM + lane * 4;
        atomic_add_f32(o + 0, m.x); atomic_add_f32(o + 1, m.y);
        atomic_add_f32(o + 2, m.z); atomic_add_f32(o + 3, m.w);
    }
    if (lane == 0) {
        atomic_add_f32(deg_f + d, 1.0f);
        atomic_add_f32(deg_r + s, 1.0f);
    }
}

// ---------------------------------------------------------------------------
// Kernel 3: fused 4x GEMM (V_WMMA_F32_16X16X4_F32) + bias + ReLU + skip.
// One block = one 16-node tile; 8 waves, each owns a 16-wide output column
// block. A operands staged in LDS (stride 132 to dodge bank conflicts);
// B operands (W^T columns) streamed from L2-resident 64 KB weight matrices.
// EXEC is all-ones through the WMMA region (no divergent exits).
// ---------------------------------------------------------------------------
__global__ void __launch_bounds__(256) gemm_kernel(
    const float* __restrict__ x,
    const float* __restrict__ h_f,   const float* __restrict__ h_r,
    const float* __restrict__ agg_f, const float* __restrict__ agg_r,
    const float* __restrict__ deg_f, const float* __restrict__ deg_r,
    const float* __restrict__ W_self,   const float* __restrict__ W_neigh,
    const float* __restrict__ b,
    const float* __restrict__ W_self_r, const float* __restrict__ W_neigh_r,
    const float* __restrict__ b_r,
    float* __restrict__ out, int n_nodes) {
    __shared__ float lds[4 * 16 * LDSS];
    float* lhf  = lds;
    float* lhnf = lds + 16 * LDSS;
    float* lhr  = lds + 2 * 16 * LDSS;
    float* lhnr = lds + 3 * 16 * LDSS;

    const int tid = threadIdx.x;
    const size_t base = (size_t)blockIdx.x * 16;

    // Cooperative stage: 4 tiles of 16x128 f32 into LDS; degree-normalize agg.
#pragma unroll
    for (int j = 0; j < 2; ++j) {
        const int idx4 = tid + j * 256;      // float4 index within tile
        const int elem = idx4 * 4;
        const int row  = elem >> 7;          // /128
        const int c    = elem & 127;
        size_t grow = base + row;
        if (grow >= (size_t)n_nodes) grow = (size_t)n_nodes - 1;  // clamp tail
        const size_t goff = grow * DIM + c;
        const int    loff = row * LDSS + c;
        float4 t;
        t = *(const float4*)(h_f + goff);
        lhf[loff + 0] = t.x; lhf[loff + 1] = t.y; lhf[loff + 2] = t.z; lhf[loff + 3] = t.w;
        const float idf = 1.0f / fmaxf(deg_f[grow], 1.0f);
        t = *(const float4*)(agg_f + goff);
        lhnf[loff + 0] = t.x * idf; lhnf[loff + 1] = t.y * idf;
        lhnf[loff + 2] = t.z * idf; lhnf[loff + 3] = t.w * idf;
        t = *(const float4*)(h_r + goff);
        lhr[loff + 0] = t.x; lhr[loff + 1] = t.y; lhr[loff + 2] = t.z; lhr[loff + 3] = t.w;
        const float idr = 1.0f / fmaxf(deg_r[grow], 1.0f);
        t = *(const float4*)(agg_r + goff);
        lhnr[loff + 0] = t.x * idr; lhnr[loff + 1] = t.y * idr;
        lhnr[loff + 2] = t.z * idr; lhnr[loff + 3] = t.w * idr;
    }
    __syncthreads();

    const int w    = tid >> 5;
    const int lane = tid & 31;
    const int n    = lane & 15;   // N index / A row (M)
    const int kh   = lane >> 4;   // K half-select per 16x16x4 A/B layout
    const int col  = w * 16 + n;  // output channel

    // B operand: B[k][n] = W[col][k]  (out = h @ W^T)
    const float* Ws   = W_self    + (size_t)col * DIM;
    const float* Wn   = W_neigh   + (size_t)col * DIM;
    const float* Wsr  = W_self_r  + (size_t)col * DIM;
    const float* Wnr  = W_neigh_r + (size_t)col * DIM;
    const float* Ahf  = lhf  + n * LDSS;
    const float* Ahnf = lhnf + n * LDSS;
    const float* Ahr  = lhr  + n * LDSS;
    const float* Ahnr = lhnr + n * LDSS;

    v8f accf = {0.f, 0.f, 0.f, 0.f, 0.f, 0.f, 0.f, 0.f};
    v8f accr = {0.f, 0.f, 0.f, 0.f, 0.f, 0.f, 0.f, 0.f};

    for (int k0 = 0; k0 < DIM; k0 += 4) {
        const int k = k0 + 2 * kh;   // lanes 0-15: K pair {0,1}; 16-31: {2,3}
        v2f a, bm;
        a  = *(const v2f*)(Ahf + k);
        bm = *(const v2f*)(Ws + k);
        accf = __builtin_amdgcn_wmma_f32_16x16x4_f32(false, a, false, bm,
                                                     (short)0, accf, false, false);
        a  = *(const v2f*)(Ahnf + k);
        bm = *(const v2f*)(Wn + k);
        accf = __builtin_amdgcn_wmma_f32_16x16x4_f32(false, a, false, bm,
                                                     (short)0, accf, false, false);
        a  = *(const v2f*)(Ahr + k);
        bm = *(const v2f*)(Wsr + k);
        accr = __builtin_amdgcn_wmma_f32_16x16x4_f32(false, a, false, bm,
                                                     (short)0, accr, false, false);
        a  = *(const v2f*)(Ahnr + k);
        bm = *(const v2f*)(Wnr + k);
        accr = __builtin_amdgcn_wmma_f32_16x16x4_f32(false, a, false, bm,
                                                     (short)0, accr, false, false);
    }

    // Epilogue per documented 16x16 f32 C/D layout:
    //   VGPR i -> row (i + 8*kh), N = lane&15.
    const float bf = b[col];
    const float br = b_r[col];
#pragma unroll
    for (int i = 0; i < 8; ++i) {
        const size_t row = base + (size_t)i + (size_t)kh * 8;
        if (row < (size_t)n_nodes) {
            const size_t idx = row * DIM + col;
            out[idx] = x[idx] + fmaxf(accf[i] + bf, 0.0f) + fmaxf(accr[i] + br, 0.0f);
        }
    }
}

// ---------------------------------------------------------------------------
extern "C" void kernel_launch(void* const* d_in, const int* in_sizes, int n_in,
                              void* d_out, int out_size, void* d_ws, size_t ws_size,
                              hipStream_t stream) {
    const float* x        = (const float*)d_in[0];
    const int*   src      = (const int*)d_in[1];
    const int*   dst      = (const int*)d_in[2];
    const float* gamma    = (const float*)d_in[3];
    const float* beta     = (const float*)d_in[4];
    const float* W_self   = (const float*)d_in[5];
    const float* W_neigh  = (const float*)d_in[6];
    const float* b        = (const float*)d_in[7];
    const float* gamma_r  = (const float*)d_in[8];
    const float* beta_r   = (const float*)d_in[9];
    const float* W_self_r = (const float*)d_in[10];
    const float* W_neigh_r= (const float*)d_in[11];
    const float* b_r      = (const float*)d_in[12];
    float* out = (float*)d_out;

    const int N = in_sizes[0] / DIM;
    const int E = in_sizes[1];

    // Workspace layout: [h_f | h_r | agg_f | agg_r | deg_f | deg_r]
    float* ws = (float*)d_ws;
    const size_t nd = (size_t)N * DIM;
    float* h_f   = ws;
    float* h_r   = ws + nd;
    float* agg_f = ws + 2 * nd;
    float* agg_r = ws + 3 * nd;
    float* deg_f = ws + 4 * nd;
    float* deg_r = deg_f + N;

    // Zero accumulators + degrees each call (graph-capturable async memset).
    hipMemsetAsync(agg_f, 0, (2 * nd + 2 * (size_t)N) * sizeof(float), stream);

    const int ln_blocks = (N + 7) / 8;     // 8 waves/block, 1 wave/node
    ln_kernel<<<ln_blocks, 256, 0, stream>>>(x, gamma, beta, gamma_r, beta_r,
                                             h_f, h_r, N);

    const int e_blocks = (E + 7) / 8;      // 1 wave/edge
    edge_kernel<<<e_blocks, 256, 0, stream>>>(src, dst, h_f, h_r,
                                              agg_f, agg_r, deg_f, deg_r, E);

    const int tiles = (N + 15) / 16;       // 1 block per 16-node tile
    gemm_kernel<<<tiles, 256, 0, stream>>>(x, h_f, h_r, agg_f, agg_r,
                                           deg_f, deg_r,
                                           W_self, W_neigh, b,
                                           W_self_r, W_neigh_r, b_r,
                                           out, N);
}